// AttentionOT_48103633715286
// MI455X (gfx1250) — compile-verified
//
#include <hip/hip_runtime.h>
#include <hip/hip_bf16.h>

typedef float v2f __attribute__((ext_vector_type(2)));
typedef float v4f __attribute__((ext_vector_type(4)));
typedef float v8f __attribute__((ext_vector_type(8)));

// D = A(16x4 f32) x B(4x16 f32) + C, full fp32 WMMA
__device__ __forceinline__ v8f wmma_f32(v2f a, v2f b, v8f c) {
    return __builtin_amdgcn_wmma_f32_16x16x4_f32(false, a, false, b, (short)0, c,
                                                 false, false);
}
__device__ __forceinline__ v2f mk2(float x, float y) { v2f r; r.x = x; r.y = y; return r; }

// ---------------------------------------------------------------------------
// Generic GEMM: Y[r,n] = sum_c X[r*C + c] * W[n*C + c]  (+ bias[n] if given)
// One wave computes a 16x64 strip (4 WMMA column tiles, A fragment reused).
// Permuted-K b128 loads: per 8-k window each lane-half h loads float4 at
// k = kb+4h; WMMA#1 uses (.x,.y), WMMA#2 uses (.z,.w) of both operands (A and
// B agree on the k->slot mapping, so the reduction is exact). 5 b128 loads per
// 8 WMMAs. Next window is prefetched, sched_barrier(0) pins the pipeline.
// Grid: (R/16)*(N/64) waves, 4 waves / block of 128 threads. Exact grids only.
// ---------------------------------------------------------------------------
__global__ void gemm_xwT(const float* __restrict__ X, const float* __restrict__ W,
                         const float* __restrict__ bias, float* __restrict__ Y,
                         int R, int C, int N) {
    const int lane = threadIdx.x & 31;
    const int w    = blockIdx.x * (blockDim.x >> 5) + (threadIdx.x >> 5);
    const int csPerRow = N >> 6;              // 64-wide column strips
    const int rt = w / csPerRow;
    const int cs = w % csPerRow;

    const int row  = rt * 16 + (lane & 15);   // A fragment row (M)
    const int h4   = (lane >> 4) * 4;         // half-wave half of the 8-k window
    const int col0 = cs * 64 + (lane & 15);   // B fragment col (N) of tile 0

    const float* xr = X + (size_t)row * C + h4;
    const float* w0 = W + (size_t)(col0 +  0) * C + h4;
    const float* w1 = W + (size_t)(col0 + 16) * C + h4;
    const float* w2 = W + (size_t)(col0 + 32) * C + h4;
    const float* w3 = W + (size_t)(col0 + 48) * C + h4;

    v8f acc[4] = {};
    v4f a  = *(const v4f*)(xr);
    v4f b0 = *(const v4f*)(w0);
    v4f b1 = *(const v4f*)(w1);
    v4f b2 = *(const v4f*)(w2);
    v4f b3 = *(const v4f*)(w3);
    for (int kb = 0; kb < C - 8; kb += 8) {
        // prefetch next 8-k window while current WMMAs execute
        v4f an  = *(const v4f*)(xr + kb + 8);
        v4f bn0 = *(const v4f*)(w0 + kb + 8);
        v4f bn1 = *(const v4f*)(w1 + kb + 8);
        v4f bn2 = *(const v4f*)(w2 + kb + 8);
        v4f bn3 = *(const v4f*)(w3 + kb + 8);
        __builtin_amdgcn_sched_barrier(0);
        const v2f al = mk2(a.x, a.y), ah = mk2(a.z, a.w);
        acc[0] = wmma_f32(al, mk2(b0.x, b0.y), acc[0]);
        acc[1] = wmma_f32(al, mk2(b1.x, b1.y), acc[1]);
        acc[2] = wmma_f32(al, mk2(b2.x, b2.y), acc[2]);
        acc[3] = wmma_f32(al, mk2(b3.x, b3.y), acc[3]);
        acc[0] = wmma_f32(ah, mk2(b0.z, b0.w), acc[0]);
        acc[1] = wmma_f32(ah, mk2(b1.z, b1.w), acc[1]);
        acc[2] = wmma_f32(ah, mk2(b2.z, b2.w), acc[2]);
        acc[3] = wmma_f32(ah, mk2(b3.z, b3.w), acc[3]);
        a = an; b0 = bn0; b1 = bn1; b2 = bn2; b3 = bn3;
    }
    {   // final window
        const v2f al = mk2(a.x, a.y), ah = mk2(a.z, a.w);
        acc[0] = wmma_f32(al, mk2(b0.x, b0.y), acc[0]);
        acc[1] = wmma_f32(al, mk2(b1.x, b1.y), acc[1]);
        acc[2] = wmma_f32(al, mk2(b2.x, b2.y), acc[2]);
        acc[3] = wmma_f32(al, mk2(b3.x, b3.y), acc[3]);
        acc[0] = wmma_f32(ah, mk2(b0.z, b0.w), acc[0]);
        acc[1] = wmma_f32(ah, mk2(b1.z, b1.w), acc[1]);
        acc[2] = wmma_f32(ah, mk2(b2.z, b2.w), acc[2]);
        acc[3] = wmma_f32(ah, mk2(b3.z, b3.w), acc[3]);
    }

    // C/D layout: VGPR r -> M = rt*16 + r + 8*(lane>=16), N = colbase + lane%16
#pragma unroll
    for (int t = 0; t < 4; ++t) {
        const int ocol = cs * 64 + t * 16 + (lane & 15);
        const float bv = bias ? bias[ocol] : 0.0f;
#pragma unroll
        for (int r = 0; r < 8; ++r) {
            const int orow = rt * 16 + r + (lane >> 4) * 8;
            Y[(size_t)orow * N + ocol] = acc[t][r] + bv;
        }
    }
}

// ---------------------------------------------------------------------------
// Row-wise L2 normalization (in-place). One wave per row.
// ---------------------------------------------------------------------------
__global__ void l2norm_rows(float* __restrict__ A, int rows, int cols) {
    const int lane = threadIdx.x & 31;
    const int w    = blockIdx.x * (blockDim.x >> 5) + (threadIdx.x >> 5);
    if (w >= rows) return;
    float* p = A + (size_t)w * cols;
    float s = 0.0f;
    for (int c = lane; c < cols; c += 32) { float x = p[c]; s += x * x; }
#pragma unroll
    for (int off = 16; off; off >>= 1) s += __shfl_xor(s, off, 32);
    const float inv = 1.0f / fmaxf(sqrtf(s), 1e-12f);
    for (int c = lane; c < cols; c += 32) p[c] *= inv;
}

// ---------------------------------------------------------------------------
// sim: S[bk, m, n] = sum_c K[b, m, c] * Q[n, kq, b, c]
//   Q flat layout: ((n*16 + kq)*8 + b)*512 + c ; K flat: (b*1024 + m)*512 + c
// One wave: fixed (bk, m-tile), computes 16 x 32 (both n tiles). 8192 waves.
// Same permuted-K b128 + prefetch + sched_barrier pipeline as gemm_xwT.
// ---------------------------------------------------------------------------
__global__ void sim_kernel(const float* __restrict__ Kn, const float* __restrict__ Qn,
                           float* __restrict__ S) {
    const int lane = threadIdx.x & 31;
    const int w    = blockIdx.x * (blockDim.x >> 5) + (threadIdx.x >> 5);
    const int bk = w >> 6;              // 64 m-tiles per bk
    const int mt = w & 63;
    const int b  = bk >> 4;
    const int kq = bk & 15;

    const int mrow = mt * 16 + (lane & 15);
    const int h4   = (lane >> 4) * 4;
    const int n0   = lane & 15;

    const float* ar = Kn + ((size_t)b * 1024 + mrow) * 512 + h4;
    const float* q0 = Qn + (((size_t)n0 * 16 + kq) * 8 + b) * 512 + h4;
    const float* q1 = Qn + (((size_t)(n0 + 16) * 16 + kq) * 8 + b) * 512 + h4;

    v8f acc0 = {}, acc1 = {};
    v4f a  = *(const v4f*)(ar);
    v4f b0 = *(const v4f*)(q0);
    v4f b1 = *(const v4f*)(q1);
    for (int kb = 0; kb < 512 - 8; kb += 8) {
        v4f an  = *(const v4f*)(ar + kb + 8);
        v4f bn0 = *(const v4f*)(q0 + kb + 8);
        v4f bn1 = *(const v4f*)(q1 + kb + 8);
        __builtin_amdgcn_sched_barrier(0);
        const v2f al = mk2(a.x, a.y), ah = mk2(a.z, a.w);
        acc0 = wmma_f32(al, mk2(b0.x, b0.y), acc0);
        acc1 = wmma_f32(al, mk2(b1.x, b1.y), acc1);
        acc0 = wmma_f32(ah, mk2(b0.z, b0.w), acc0);
        acc1 = wmma_f32(ah, mk2(b1.z, b1.w), acc1);
        a = an; b0 = bn0; b1 = bn1;
    }
    {
        const v2f al = mk2(a.x, a.y), ah = mk2(a.z, a.w);
        acc0 = wmma_f32(al, mk2(b0.x, b0.y), acc0);
        acc1 = wmma_f32(al, mk2(b1.x, b1.y), acc1);
        acc0 = wmma_f32(ah, mk2(b0.z, b0.w), acc0);
        acc1 = wmma_f32(ah, mk2(b1.z, b1.w), acc1);
    }

    float* out = S + (size_t)bk * 1024 * 32;
#pragma unroll
    for (int r = 0; r < 8; ++r) {
        const int m = mt * 16 + r + (lane >> 4) * 8;
        const int n = lane & 15;
        out[(size_t)m * 32 + n]      = acc0[r];
        out[(size_t)m * 32 + n + 16] = acc1[r];
    }
}

// ---------------------------------------------------------------------------
// Log-domain Sinkhorn, 100 iterations, one workgroup per bk slice.
// The sim slice (raw, NOT 1-sim) is staged global->LDS with CDNA5 async
// copies (GLOBAL_LOAD_ASYNC_TO_LDS_B32 + s_wait_asynccnt); the constant -1 of
// wdist = 1 - sim is folded into the updates: LSE((x-1)/eps) = -1/eps + LSE(x/eps)
//   u = eps*logmu + 1 - eps*LSE_n((sim+v)/eps)
//   v = eps*lognu + 1 - eps*LSE_m((sim+u)/eps)
//   T = exp((sim - 1 + u + v)/eps)
// Writes T (overwrites S in place) and attn_save = M*Nq*sum_n sim*T.
// LDS: 1024x33 padded sim (135168B) + u + v + scratch = 140.5 KB (<320KB/WGP).
// ---------------------------------------------------------------------------
#define SINK_LDS_FLOATS (1024 * 33 + 1024 + 32 + 256 + 32)

__global__ void sinkhorn_kernel(float* __restrict__ S, float* __restrict__ attn_save) {
    extern __shared__ float smem[];
    float* Cm   = smem;                 // 1024*33 padded sim matrix
    float* u_s  = Cm + 1024 * 33;       // 1024
    float* v_s  = u_s + 1024;           // 32
    float* red  = v_s + 32;             // 256 reduction scratch
    float* colr = red + 256;            // 32 per-column combined values

    const int tid = threadIdx.x;        // blockDim = 256
    const int bk  = blockIdx.x;
    const float eps = 0.05f, inv_eps = 20.0f;
    const float logmu = __logf(1.0f / 1024.0f + 1e-8f);
    const float lognu = __logf(1.0f / 32.0f + 1e-8f);

    float* Sbk = S + (size_t)bk * 32768;

    // ---- async global -> LDS staging of the sim slice (pad 33 per row) ----
    {
        const unsigned lds_base = __builtin_amdgcn_groupstaticsize();
        for (int idx = tid; idx < 32768; idx += 256) {
            const unsigned lds_off =
                lds_base + (unsigned)(((idx >> 5) * 33 + (idx & 31)) * 4);
            const unsigned long long ga =
                (unsigned long long)(uintptr_t)(Sbk + idx);
            asm volatile("global_load_async_to_lds_b32 %0, %1, off"
                         :: "v"(lds_off), "v"(ga) : "memory");
        }
        asm volatile("s_wait_asynccnt 0x0" ::: "memory");
    }
    if (tid < 32) v_s[tid] = 0.0f;
    __syncthreads();

    for (int it = 0; it < 100; ++it) {
        // ---- u update: per-row LSE over n (each thread owns 4 rows) ----
        for (int j = 0; j < 4; ++j) {
            const int m = tid + j * 256;
            const float* row = Cm + m * 33;
            float mx = -3.0e38f;
#pragma unroll
            for (int n = 0; n < 32; ++n) mx = fmaxf(mx, v_s[n] + row[n]);
            float ss = 0.0f;
#pragma unroll
            for (int n = 0; n < 32; ++n)
                ss += __expf((v_s[n] + row[n] - mx) * inv_eps);
            u_s[m] = eps * logmu + 1.0f - mx - eps * __logf(ss);
        }
        __syncthreads();
        // ---- v update: per-column LSE over m (8 threads per column) ----
        {
            const int n = tid >> 3, sub = tid & 7;
            float mx = -3.0e38f;
            for (int m = sub; m < 1024; m += 8)
                mx = fmaxf(mx, u_s[m] + Cm[m * 33 + n]);
            red[tid] = mx;
            __syncthreads();
            if (sub == 0) {
                float m2 = red[tid];
                for (int i = 1; i < 8; ++i) m2 = fmaxf(m2, red[tid + i]);
                colr[n] = m2;
            }
            __syncthreads();
            mx = colr[n];
            float ss = 0.0f;
            for (int m = sub; m < 1024; m += 8)
                ss += __expf((u_s[m] + Cm[m * 33 + n] - mx) * inv_eps);
            red[tid] = ss;
            __syncthreads();
            if (sub == 0) {
                float s2 = 0.0f;
                for (int i = 0; i < 8; ++i) s2 += red[tid + i];
                v_s[n] = eps * lognu + 1.0f - mx - eps * __logf(s2);
            }
            __syncthreads();
        }
    }

    // ---- finalize: T = exp((sim-1+u+v)/eps), attn_save = 32768*sum_n sim*T ----
    for (int j = 0; j < 4; ++j) {
        const int m = tid + j * 256;
        const float um1 = u_s[m] - 1.0f;
        const float* row = Cm + m * 33;
        float s = 0.0f;
#pragma unroll
        for (int n = 0; n < 32; ++n) {
            const float smn = row[n];
            const float Tmn = __expf((um1 + v_s[n] + smn) * inv_eps);
            Sbk[(size_t)m * 32 + n] = Tmn;          // overwrite sim with T
            s += smn * Tmn;
        }
        attn_save[(size_t)bk * 1024 + m] = 32768.0f * s;   // M*Nq = 32768
    }
}

// ---------------------------------------------------------------------------
// x_pre[kq, n, b, c] = sum_m T[bk, m, n] * V[b, m, c]
// Per wave: fixed (bk, n-tile of 16, c-strip of 64). GEMM 32 x 512 x 1024.
// Both operands are k(m)-strided in memory, so fragments use scalar loads;
// an 8-deep m-window is prefetched and pinned with sched_barrier(0).
// 128 * 2 * 8 = 2048 waves.
// ---------------------------------------------------------------------------
__global__ void attnv_kernel(const float* __restrict__ T, const float* __restrict__ V,
                             float* __restrict__ Xp) {
    const int lane = threadIdx.x & 31;
    const int w    = blockIdx.x * (blockDim.x >> 5) + (threadIdx.x >> 5);
    const int cs = w & 7;
    const int nt = (w >> 3) & 1;
    const int bk = w >> 4;
    const int b  = bk >> 4;
    const int kq = bk & 15;

    const int koff = (lane >> 4) * 2;
    const int nf   = nt * 16 + (lane & 15);      // A row = n
    const int c0   = cs * 64 + (lane & 15);
    const float* Tb = T + (size_t)bk * 32768;    // T[m*32 + n]
    const float* Vb = V + (size_t)b * 1024 * 512;

    auto ldA = [&](int m0) {
        return mk2(Tb[(size_t)m0 * 32 + nf], Tb[(size_t)(m0 + 1) * 32 + nf]);
    };
    auto ldB = [&](int m0, int t) {
        const int c = c0 + t * 16;
        return mk2(Vb[(size_t)m0 * 512 + c], Vb[(size_t)(m0 + 1) * 512 + c]);
    };

    v8f acc[4] = {};
    v2f a[2], bb[2][4];
#pragma unroll
    for (int s = 0; s < 2; ++s) {
        const int m0 = 4 * s + koff;
        a[s] = ldA(m0);
#pragma unroll
        for (int t = 0; t < 4; ++t) bb[s][t] = ldB(m0, t);
    }
    for (int kb = 0; kb < 1024 - 8; kb += 8) {
        v2f an[2], bn[2][4];
#pragma unroll
        for (int s = 0; s < 2; ++s) {
            const int m0 = kb + 8 + 4 * s + koff;
            an[s] = ldA(m0);
#pragma unroll
            for (int t = 0; t < 4; ++t) bn[s][t] = ldB(m0, t);
        }
        __builtin_amdgcn_sched_barrier(0);
#pragma unroll
        for (int s = 0; s < 2; ++s)
#pragma unroll
            for (int t = 0; t < 4; ++t)
                acc[t] = wmma_f32(a[s], bb[s][t], acc[t]);
#pragma unroll
        for (int s = 0; s < 2; ++s) {
            a[s] = an[s];
#pragma unroll
            for (int t = 0; t < 4; ++t) bb[s][t] = bn[s][t];
        }
    }
#pragma unroll
    for (int s = 0; s < 2; ++s)
#pragma unroll
        for (int t = 0; t < 4; ++t)
            acc[t] = wmma_f32(a[s], bb[s][t], acc[t]);

#pragma unroll
    for (int t = 0; t < 4; ++t) {
#pragma unroll
        for (int r = 0; r < 8; ++r) {
            const int n = nt * 16 + r + (lane >> 4) * 8;
            const int c = cs * 64 + t * 16 + (lane & 15);
            Xp[(((size_t)kq * 32 + n) * 8 + b) * 512 + c] = acc[t][r];
        }
    }
}

// ---------------------------------------------------------------------------
// Host-side orchestration.
// Inputs: xq[32,16,8,512] xk[8,1024,512] xv[8,1024,512] Wq Wk Wv Wp [512,512] bp[512]
// Output: x[16,32,8,512] (2097152 floats) ++ attn_save[8,16,1024] (131072 floats)
// Workspace layout (floats): q 2M | k 4M | v 4M | S/T 4M | xpre 2M  = 64 MB
// ---------------------------------------------------------------------------
extern "C" void kernel_launch(void* const* d_in, const int* in_sizes, int n_in,
                              void* d_out, int out_size, void* d_ws, size_t ws_size,
                              hipStream_t stream) {
    const float* xq = (const float*)d_in[0];
    const float* xk = (const float*)d_in[1];
    const float* xv = (const float*)d_in[2];
    const float* Wq = (const float*)d_in[3];
    const float* Wk = (const float*)d_in[4];
    const float* Wv = (const float*)d_in[5];
    const float* Wp = (const float*)d_in[6];
    const float* bp = (const float*)d_in[7];

    float* out = (float*)d_out;
    float* ws  = (float*)d_ws;
    float* q   = ws;                  // 4096 x 512
    float* kn  = ws + 2097152;        // 8192 x 512
    float* vn  = ws + 6291456;        // 8192 x 512
    float* S   = ws + 10485760;       // 128 x 1024 x 32 (sim, then T in place)
    float* xp  = ws + 14680064;       // 4096 x 512

    // q/k/v projections (no bias)
    gemm_xwT<<<512, 128, 0, stream>>>(xq, Wq, nullptr, q, 4096, 512, 512);
    gemm_xwT<<<1024, 128, 0, stream>>>(xk, Wk, nullptr, kn, 8192, 512, 512);
    gemm_xwT<<<1024, 128, 0, stream>>>(xv, Wv, nullptr, vn, 8192, 512, 512);

    // l2 normalize q and k along C
    l2norm_rows<<<512, 256, 0, stream>>>(q, 4096, 512);
    l2norm_rows<<<1024, 256, 0, stream>>>(kn, 8192, 512);

    // sim[bk, m, n]
    sim_kernel<<<2048, 128, 0, stream>>>(kn, q, S);

    // Sinkhorn (needs 140.5 KB dynamic LDS; CDNA5 WGP has 320 KB)
    const int sink_lds = SINK_LDS_FLOATS * (int)sizeof(float);
    (void)hipFuncSetAttribute(reinterpret_cast<const void*>(sinkhorn_kernel),
                              hipFuncAttributeMaxDynamicSharedMemorySize, sink_lds);
    sinkhorn_kernel<<<128, 256, sink_lds, stream>>>(S, out + 2097152);

    // x_pre = T^T @ V  per (b,kq)
    attnv_kernel<<<512, 128, 0, stream>>>(S, vn, xp);

    // final projection with bias, directly into d_out
    gemm_xwT<<<512, 128, 0, stream>>>(xp, Wp, bp, out, 4096, 512, 512);
}